// WSDAccuracyCosineSimilarity_66176856096810
// MI455X (gfx1250) — compile-verified
//
#include <hip/hip_runtime.h>
#include <hip/hip_bf16.h>
#include <math.h>

// Shapes from the reference
#define BB 32
#define SS 256
#define KK 32
#define DD 300
#define NTOK (BB * SS)        // 8192 tokens
#define WPB 8                 // waves (tokens) per block; 256 threads
#define NBLK (NTOK / WPB)     // 1024 blocks
#define EPSF 1e-12f

typedef __attribute__((ext_vector_type(2))) float v2f;
typedef __attribute__((ext_vector_type(8))) float v8f;

// One wave32 per token. WMMA f32 16x16x4 accumulates the 32 candidate dot
// products over D=300 in 75 chunks of 4. B matrix is the pred chunk broadcast
// to all 16 columns, so every accumulator column holds the same dot value.
__global__ void __launch_bounds__(WPB * 32)
wsd_token_kernel(const float* __restrict__ pred,      // [NTOK, D]
                 const float* __restrict__ cands,     // [NTOK, K, D]
                 const int* __restrict__ truep,       // [NTOK]
                 const float* __restrict__ sweight,   // [NTOK]
                 float* __restrict__ partials)        // [NBLK, 2]
{
    const int lane = threadIdx.x & 31;
    const int wave = threadIdx.x >> 5;
    const int token = blockIdx.x * WPB + wave;

    const float* p = pred  + (size_t)token * DD;
    const float* c = cands + (size_t)token * KK * DD;

    // A-matrix 16x4 f32 layout: lanes 0-15 -> rows M=0..15, VGPR0/1 = K0/K1;
    // lanes 16-31 -> same rows, VGPR0/1 = K2/K3. So each lane loads 2
    // consecutive floats at column offset (lane>>4)*2. B (4x16) uses the same
    // K mapping, independent of column -> same addressing from pred.
    const int half = lane >> 4;
    const int row  = lane & 15;
    const int off  = half * 2;

    const float* rowp0 = c + (size_t)row * DD + off;          // candidates 0..15
    const float* rowp1 = c + (size_t)(row + 16) * DD + off;   // candidates 16..31
    const float* pp    = p + off;

    v8f acc0 = {};  // dots for candidates 0..15 (broadcast across columns)
    v8f acc1 = {};  // dots for candidates 16..31
    float n0 = 0.f, n1 = 0.f;   // per-lane partial squared-norms

    #pragma unroll 5
    for (int d = 0; d < DD; d += 4) {
        v2f a0 = *(const v2f*)(rowp0 + d);
        v2f a1 = *(const v2f*)(rowp1 + d);
        v2f bv = *(const v2f*)(pp + d);
        acc0 = __builtin_amdgcn_wmma_f32_16x16x4_f32(
                   false, a0, false, bv, (short)0, acc0, false, false);
        acc1 = __builtin_amdgcn_wmma_f32_16x16x4_f32(
                   false, a1, false, bv, (short)0, acc1, false, false);
        n0 += a0.x * a0.x + a0.y * a0.y;
        n1 += a1.x * a1.x + a1.y * a1.y;
    }

    __shared__ float s_dot[WPB][32];
    __shared__ float s_n0[WPB][32];
    __shared__ float s_n1[WPB][32];
    __shared__ float s_cw[WPB];
    __shared__ float s_w[WPB];

    s_n0[wave][lane] = n0;
    s_n1[wave][lane] = n1;

    // C/D layout: VGPR r, lane l -> row M = r + 8*(l>>4); all columns equal.
    if (lane == 0) {
        #pragma unroll
        for (int r = 0; r < 8; ++r) {
            s_dot[wave][r]      = acc0[r];   // candidates 0..7
            s_dot[wave][16 + r] = acc1[r];   // candidates 16..23
        }
    }
    if (lane == 16) {
        #pragma unroll
        for (int r = 0; r < 8; ++r) {
            s_dot[wave][8 + r]  = acc0[r];   // candidates 8..15
            s_dot[wave][24 + r] = acc1[r];   // candidates 24..31
        }
    }
    __syncthreads();

    // Lane k scores candidate k. |pred| scaling is a common positive factor
    // and cannot change the argmax, so it is omitted.
    float dk = s_dot[wave][lane];
    float n2 = (lane < 16) ? (s_n0[wave][lane] + s_n0[wave][lane + 16])
                           : (s_n1[wave][lane - 16] + s_n1[wave][lane]);
    float score = -dk / fmaxf(sqrtf(n2), EPSF);

    // Wave32 argmax with first-index tie-break (matches jnp.argmax).
    float bvv = score;
    int   bii = lane;
    #pragma unroll
    for (int o = 16; o > 0; o >>= 1) {
        float ov = __shfl_down(bvv, o, 32);
        int   oi = __shfl_down(bii, o, 32);
        if (ov > bvv || (ov == bvv && oi < bii)) { bvv = ov; bii = oi; }
    }

    if (lane == 0) {
        float w = sweight[token];
        s_cw[wave] = (bii == truep[token]) ? w : 0.f;
        s_w[wave]  = w;
    }
    __syncthreads();

    if (threadIdx.x == 0) {
        float cw = 0.f, ww = 0.f;
        #pragma unroll
        for (int i = 0; i < WPB; ++i) { cw += s_cw[i]; ww += s_w[i]; }
        partials[(size_t)blockIdx.x * 2]     = cw;   // unconditional overwrite
        partials[(size_t)blockIdx.x * 2 + 1] = ww;
    }
}

// Deterministic single-block final reduction over NBLK partial pairs.
__global__ void __launch_bounds__(256)
wsd_reduce_kernel(const float* __restrict__ partials, float* __restrict__ out)
{
    __shared__ float scw[256];
    __shared__ float sw[256];
    const int t = threadIdx.x;
    float cw = 0.f, ww = 0.f;
    for (int i = t; i < NBLK; i += 256) {
        cw += partials[(size_t)i * 2];
        ww += partials[(size_t)i * 2 + 1];
    }
    scw[t] = cw;
    sw[t]  = ww;
    __syncthreads();
    for (int s = 128; s > 0; s >>= 1) {
        if (t < s) { scw[t] += scw[t + s]; sw[t] += sw[t + s]; }
        __syncthreads();
    }
    if (t == 0) out[0] = scw[0] / fmaxf(sw[0], EPSF);
}

extern "C" void kernel_launch(void* const* d_in, const int* in_sizes, int n_in,
                              void* d_out, int out_size, void* d_ws, size_t ws_size,
                              hipStream_t stream) {
    const float* pred    = (const float*)d_in[0];  // [B,S,D] f32
    const float* cands   = (const float*)d_in[1];  // [B,S,K,D] f32
    const int*   truep   = (const int*)d_in[2];    // [B,S] i32
    const float* sweight = (const float*)d_in[3];  // [B,S] f32
    float* out      = (float*)d_out;               // scalar f32
    float* partials = (float*)d_ws;                // NBLK*2 floats (8 KB)

    wsd_token_kernel<<<NBLK, WPB * 32, 0, stream>>>(pred, cands, truep, sweight, partials);
    wsd_reduce_kernel<<<1, 256, 0, stream>>>(partials, out);
}